// GHMC_Loss_9895604649991
// MI455X (gfx1250) — compile-verified
//
#include <hip/hip_runtime.h>
#include <hip/hip_bf16.h>
#include <stdint.h>

#define BINS    30
#define BLOCK   256
#define WAVES   (BLOCK / 32)
#define TILE    8                 // elements per thread per tile (2 x float4)
#define NBUF    3                 // async pipeline depth (prefetch distance 2)
#define HSTRIDE 33                // pad to spread LDS banks

// ---- CDNA5 async global -> LDS copy (ASYNCcnt path) -----------------------
__device__ __forceinline__ void async_copy_b128(uint32_t lds_byte_addr,
                                                const void* gaddr) {
  // GV mode: vdst = LDS byte address, vaddr = 64-bit global address
  asm volatile("global_load_async_to_lds_b128 %0, %1, off"
               :
               : "v"(lds_byte_addr), "v"(gaddr)
               : "memory");
}
#define WAIT_ASYNCCNT(n) asm volatile("s_wait_asynccnt " #n ::: "memory")

__device__ __forceinline__ uint32_t lds_addr_of(const void* p) {
  // generic pointer to LDS: low 32 bits are the wave-relative LDS byte offset
  return (uint32_t)(uintptr_t)p;
}

// ---------------------------------------------------------------------------
__device__ __forceinline__ void accum_elem(float x, float t,
                                           float* hs, unsigned* hc) {
  float e   = __expf(-x);                          // v_exp_f32
  float sig = __builtin_amdgcn_rcpf(1.0f + e);     // v_rcp_f32
  float g   = fabsf(sig - t);
  int   b   = (int)(g * (float)BINS);              // trunc toward zero, g>=0
  b = b < 0 ? 0 : (b > BINS - 1 ? BINS - 1 : b);
  float ea  = __expf(-fabsf(x));                   // v_exp_f32
  float bce = fmaxf(x, 0.0f) - x * t + __logf(1.0f + ea);  // stable softplus-x*t
  atomicAdd(&hs[b], bce);                          // ds_add_f32
  atomicAdd(&hc[b], 1u);                           // ds_add_u32
}

__global__ void ghm_init_kernel(unsigned* gCnt, float* gSum) {
  int t = threadIdx.x;
  if (t < BINS) { gCnt[t] = 0u; gSum[t] = 0.0f; }
}

__global__ __launch_bounds__(BLOCK)
void ghm_partial_kernel(const float* __restrict__ in,
                        const int* __restrict__ tg,
                        int n,
                        unsigned* __restrict__ gCnt,
                        float* __restrict__ gSum,
                        int sweeps, int stride) {
  __shared__ float    sIn[NBUF][BLOCK * TILE];   // 3 x 8KB staging
  __shared__ int      sTg[NBUF][BLOCK * TILE];   // 3 x 8KB staging
  __shared__ float    hSum[WAVES][HSTRIDE];      // per-wave bin BCE sums
  __shared__ unsigned hCnt[WAVES][HSTRIDE];      // per-wave bin counts

  const int tid  = threadIdx.x;
  const int w    = tid >> 5;
  const int lane = tid & 31;
  hSum[w][lane] = 0.0f;        // each wave zeroes its own row (DS in-order)
  hCnt[w][lane] = 0u;

  const long long gbase = ((long long)blockIdx.x * BLOCK + tid) * TILE;
  const long long step  = (long long)stride;

  uint32_t ldsIn[NBUF], ldsTg[NBUF];
  #pragma unroll
  for (int b = 0; b < NBUF; ++b) {
    ldsIn[b] = lds_addr_of(&sIn[b][tid * TILE]);
    ldsTg[b] = lds_addr_of(&sTg[b][tid * TILE]);
  }

  auto clamped = [&](int s) -> long long {
    long long i = gbase + (long long)s * step;
    if (i > (long long)n - TILE) i = (long long)n - TILE;  // uniform pipeline
    return i;
  };
  auto stage = [&](int s, int buf) {                       // 4 async B128 / tile
    long long i = clamped(s);
    async_copy_b128(ldsIn[buf],      in + i);
    async_copy_b128(ldsIn[buf] + 16, in + i + 4);
    async_copy_b128(ldsTg[buf],      tg + i);
    async_copy_b128(ldsTg[buf] + 16, tg + i + 4);
  };

  // prologue: stage tiles 0 and 1
  stage(0, 0);
  if (sweeps > 1) stage(1, 1);

  int buf = 0;
  for (int s = 0; s < sweeps; ++s) {
    if (s + 2 < sweeps) {
      int nb = buf + 2; if (nb >= NBUF) nb -= NBUF;
      stage(s + 2, nb);
      WAIT_ASYNCCNT(8);        // tiles s+1, s+2 may stay in flight; s landed
    } else if (s + 1 < sweeps) {
      WAIT_ASYNCCNT(4);        // tile s+1 may stay in flight
    } else {
      WAIT_ASYNCCNT(0);        // drain
    }

    const long long idx = gbase + (long long)s * step;
    if (idx + TILE <= (long long)n) {
      const float* pi = &sIn[buf][tid * TILE];
      const int*   pt = &sTg[buf][tid * TILE];
      float4 x4a = *(const float4*)(pi);         // ds_load_b128
      float4 x4b = *(const float4*)(pi + 4);     // ds_load_b128
      int4   t4a = *(const int4*)(pt);           // ds_load_b128
      int4   t4b = *(const int4*)(pt + 4);       // ds_load_b128
      accum_elem(x4a.x, (float)t4a.x, hSum[w], hCnt[w]);
      accum_elem(x4a.y, (float)t4a.y, hSum[w], hCnt[w]);
      accum_elem(x4a.z, (float)t4a.z, hSum[w], hCnt[w]);
      accum_elem(x4a.w, (float)t4a.w, hSum[w], hCnt[w]);
      accum_elem(x4b.x, (float)t4b.x, hSum[w], hCnt[w]);
      accum_elem(x4b.y, (float)t4b.y, hSum[w], hCnt[w]);
      accum_elem(x4b.z, (float)t4b.z, hSum[w], hCnt[w]);
      accum_elem(x4b.w, (float)t4b.w, hSum[w], hCnt[w]);
    } else if (idx < (long long)n) {
      // ragged tail (not hit for the harness shape): direct global loads
      for (int k = 0; k < TILE; ++k) {
        long long j = idx + k;
        if (j < (long long)n)
          accum_elem(in[j], (float)tg[j], hSum[w], hCnt[w]);
      }
    }
    ++buf; if (buf >= NBUF) buf = 0;
  }

  __syncthreads();

  // fold per-wave histograms and flush to global accumulators
  if (tid < BINS) {
    float    s = 0.0f;
    unsigned c = 0u;
    #pragma unroll
    for (int i = 0; i < WAVES; ++i) { s += hSum[i][tid]; c += hCnt[i][tid]; }
    atomicAdd(&gSum[tid], s);   // global_atomic_add_f32
    atomicAdd(&gCnt[tid], c);   // global_atomic_add_u32 (exact counts)
  }
}

__global__ void ghm_final_kernel(const unsigned* __restrict__ gCnt,
                                 const float* __restrict__ gSum,
                                 float* __restrict__ out, float tot) {
  if (threadIdx.x == 0 && blockIdx.x == 0) {
    float nvalid = 0.0f;
    for (int b = 0; b < BINS; ++b)
      if (gCnt[b] > 0u) nvalid += 1.0f;
    nvalid = fmaxf(nvalid, 1.0f);
    float total = 0.0f;
    for (int b = 0; b < BINS; ++b) {
      if (gCnt[b] > 0u) {
        float acc = 0.5f * (float)gCnt[b];                 // (1-momentum)*count
        float bw  = tot / fmaxf(acc, 1e-12f);              // bin weight
        total += gSum[b] * (bw / nvalid);
      }
    }
    out[0] = total / tot;
  }
}

// ---------------------------------------------------------------------------
extern "C" void kernel_launch(void* const* d_in, const int* in_sizes, int n_in,
                              void* d_out, int out_size, void* d_ws, size_t ws_size,
                              hipStream_t stream) {
  const float* in = (const float*)d_in[0];
  const int*   tg = (const int*)d_in[1];
  const int    n  = in_sizes[0];            // 8192*4096 = 33,554,432

  unsigned* gCnt = (unsigned*)d_ws;         // 30 u32 counts
  float*    gSum = (float*)d_ws + 64;       // 30 f32 BCE sums (256B offset)

  ghm_init_kernel<<<1, 64, 0, stream>>>(gCnt, gSum);

  const int blocks = 2048;
  const int stride = blocks * BLOCK * TILE; // elements per sweep = 4,194,304
  const int sweeps = (n + stride - 1) / stride;

  ghm_partial_kernel<<<blocks, BLOCK, 0, stream>>>(in, tg, n, gCnt, gSum,
                                                   sweeps, stride);

  ghm_final_kernel<<<1, 32, 0, stream>>>(gCnt, gSum, (float*)d_out, (float)n);
}